// GNNRLAgent_41162966565411
// MI455X (gfx1250) — compile-verified
//
#include <hip/hip_runtime.h>

#define BB 4
#define NN 50000
#define EE 160000
#define KK 1000
#define CC 128

typedef float v2f __attribute__((ext_vector_type(2)));
typedef float v8f __attribute__((ext_vector_type(8)));
typedef unsigned int u32x4 __attribute__((ext_vector_type(4)));
typedef int i32x8 __attribute__((ext_vector_type(8)));
typedef int i32x4 __attribute__((ext_vector_type(4)));

// fp32 WMMA: D(16x16,f32) = A(16x4,f32) * B(4x16,f32) + C
// A layout: lanes 0-15 row M=lane, elems {K, K+1}; lanes 16-31 elems {K+2, K+3}
// D layout: vgpr v -> M = v + 8*(lane>>4), N = lane&15
__device__ __forceinline__ v8f wmma_f32(v2f a, v2f b, v8f c) {
  return __builtin_amdgcn_wmma_f32_16x16x4_f32(false, a, false, b, (short)0, c,
                                               false, false);
}

#if __has_builtin(__builtin_amdgcn_tensor_load_to_lds)
#define HAVE_TDM 1
// 2-D TDM copy: tile 128 cols x 128 rows of f32 from row-major global into
// LDS at lds_off. tensor_dim1 = remRows => rows past the end are zero-filled
// by TDM out-of-bounds semantics (handles the ragged tail with no branches).
__device__ __forceinline__ void tdm_load_tile(const float* gsrc,
                                              unsigned lds_off, int remRows) {
  unsigned long long ga = (unsigned long long)(uintptr_t)gsrc;
  u32x4 g0;
  g0.x = 1u;                                    // count=1 (valid user D#)
  g0.y = lds_off;                               // lds_addr (bytes)
  g0.z = (unsigned)(ga & 0xFFFFFFFFu);          // global_addr[31:0]
  g0.w = ((unsigned)(ga >> 32) & 0x01FFFFFFu)   // global_addr[56:32]
         | 0x80000000u;                         // type=2 ("image")
  i32x8 g1;
  g1[0] = 0x00020000;                           // data_size=2 -> 4 bytes
  g1[1] = (int)(128u << 16);                    // tensor_dim0=128 (lo16)
  g1[2] = (int)(((unsigned)remRows & 0xFFFFu) << 16);  // tensor_dim1 lo16
  g1[3] = (int)((((unsigned)remRows >> 16) & 0xFFFFu)  // tensor_dim1 hi16
                | (128u << 16));                // tile_dim0=128
  g1[4] = 128;                                  // tile_dim1=128, tile_dim2=0
  g1[5] = 128;                                  // tensor_dim0_stride lo32
  g1[6] = 0;
  g1[7] = 0;
  i32x4 z4 = {0, 0, 0, 0};
  i32x8 z8 = {0, 0, 0, 0, 0, 0, 0, 0};
  __builtin_amdgcn_tensor_load_to_lds(g0, g1, z4, z4, z8, 0);
}
#else
#define HAVE_TDM 0
#endif

// ---------------- Layer-1 scalar segment-sum -----------------------------
__global__ __launch_bounds__(256) void scatter0_kernel(
    const float* __restrict__ x, const int* __restrict__ edges,
    float* __restrict__ agg0) {
  int tid = blockIdx.x * 256 + threadIdx.x;
  if (tid >= BB * EE) return;
  int b = tid / EE, e = tid - b * EE;
  const int* eb = edges + (size_t)b * 2 * EE;
  int src = eb[e];
  int dst = eb[EE + e];
  float v = x[(size_t)b * NN + src];
  __hip_atomic_fetch_add(&agg0[(size_t)b * NN + dst], v, __ATOMIC_RELAXED,
                         __HIP_MEMORY_SCOPE_AGENT);
}

// ---------------- t0 = relu((x+agg0) * w1_first + b1[0]) ------------------
__global__ __launch_bounds__(256) void t0_kernel(
    const float* __restrict__ x, const float* __restrict__ agg0,
    const float* __restrict__ w1f, const float* __restrict__ b1_0,
    float* __restrict__ T) {
  size_t tid = (size_t)blockIdx.x * 256 + threadIdx.x;
  if (tid >= (size_t)BB * NN * 32) return;
  int c0 = (int)(tid & 31) * 4;
  size_t rn = tid >> 5;  // flat (b*N + n)
  float z = x[rn] + agg0[rn];
  float4 w = *(const float4*)&w1f[c0];
  float4 bb = *(const float4*)&b1_0[c0];
  float4 o;
  o.x = fmaxf(z * w.x + bb.x, 0.f);
  o.y = fmaxf(z * w.y + bb.y, 0.f);
  o.z = fmaxf(z * w.z + bb.z, 0.f);
  o.w = fmaxf(z * w.w + bb.w, 0.f);
  *(float4*)&T[rn * CC + c0] = o;
}

// ---------------- vector segment-sum: A[dst] += H[src] --------------------
__global__ __launch_bounds__(256) void scatter_kernel(
    const float* __restrict__ H, const int* __restrict__ edges,
    float* __restrict__ A) {
  size_t tid = (size_t)blockIdx.x * 256 + threadIdx.x;
  if (tid >= (size_t)BB * EE * 32) return;
  int c0 = (int)(tid & 31) * 4;
  size_t be = tid >> 5;
  int b = (int)(be / EE), e = (int)(be - (size_t)b * EE);
  const int* eb = edges + (size_t)b * 2 * EE;
  int src = eb[e];
  int dst = eb[EE + e];
  float4 v = *(const float4*)&H[((size_t)b * NN + src) * CC + c0];
  float* ap = &A[((size_t)b * NN + dst) * CC + c0];
  __hip_atomic_fetch_add(ap + 0, v.x, __ATOMIC_RELAXED, __HIP_MEMORY_SCOPE_AGENT);
  __hip_atomic_fetch_add(ap + 1, v.y, __ATOMIC_RELAXED, __HIP_MEMORY_SCOPE_AGENT);
  __hip_atomic_fetch_add(ap + 2, v.z, __ATOMIC_RELAXED, __HIP_MEMORY_SCOPE_AGENT);
  __hip_atomic_fetch_add(ap + 3, v.w, __ATOMIC_RELAXED, __HIP_MEMORY_SCOPE_AGENT);
}

// ---------------- out = relu((in1[+in2]) @ W + bias) via fp32 WMMA --------
// grid: (ceil(rows/128), B); 4 waves/block, each wave owns 32 rows x 128 cols.
// A-tile staged to LDS by the Tensor Data Mover (zero-filled ragged tail);
// weights staged in a (k,k+1)-paired LDS layout so each B-frag is one b64.
template <bool HAS_IN2>
__global__ __launch_bounds__(128) void gemm_kernel(
    const float* __restrict__ in1, const float* __restrict__ in2,
    const float* __restrict__ W, const float* __restrict__ bias,
    float* __restrict__ out, int rows) {
  __shared__ __align__(16) float sW[CC * CC];   // 64KB paired weights
  __shared__ __align__(16) float sA[128 * CC];  // 64KB TDM-staged A tile
  int b = blockIdx.y;
  int rowBase = blockIdx.x * 128;
  const float* i1 = in1 + (size_t)b * NN * CC;
  const float* i2 = HAS_IN2 ? in2 + (size_t)b * NN * CC : nullptr;
  float* o = out + (size_t)b * NN * CC;

#if HAVE_TDM
  if (threadIdx.x < 32) {
    tdm_load_tile(i1 + (size_t)rowBase * CC, (unsigned)(uintptr_t)sA,
                  rows - rowBase);
  }
#endif

  // stage W: sW[(k/2)*256 + n*2 + (k&1)] = W[k*128+n]
  for (int i = threadIdx.x; i < CC * CC; i += 128) {
    int k = i >> 7, n = i & 127;
    sW[(k >> 1) * 256 + n * 2 + (k & 1)] = W[i];
  }

#if HAVE_TDM
  if (threadIdx.x < 32) __builtin_amdgcn_s_wait_tensorcnt(0);
#else
  {
    // fallback: cooperative load of the A tile with bounds clamp
    for (int i = threadIdx.x; i < 128 * (CC / 4); i += 128) {
      int r = i / (CC / 4), c4 = (i % (CC / 4)) * 4;
      int gr = rowBase + r;
      float4 v = make_float4(0.f, 0.f, 0.f, 0.f);
      if (gr < rows) v = *(const float4*)&i1[(size_t)gr * CC + c4];
      *(float4*)&sA[r * CC + c4] = v;
    }
  }
#endif
  __syncthreads();

  int wave = threadIdx.x >> 5, lane = threadIdx.x & 31;
  int lo = lane & 15, hi = lane >> 4;
  int lr0 = wave * 32 + lo, lr1 = lr0 + 16;
  int gr0 = rowBase + lr0, gr1 = rowBase + lr1;
  int cr0 = gr0 < rows ? gr0 : rows - 1;  // clamp (adds garbage only to
  int cr1 = gr1 < rows ? gr1 : rows - 1;  //  rows whose stores are masked)
  const v2f* i2v = (const v2f*)i2;

  v8f acc0[8] = {};
  v8f acc1[8] = {};
#pragma unroll 4
  for (int k = 0; k < CC; k += 4) {
    int kk = k + 2 * hi;
    v2f a0 = *(const v2f*)&sA[lr0 * CC + kk];
    v2f a1 = *(const v2f*)&sA[lr1 * CC + kk];
    if (HAS_IN2) {
      a0 += i2v[(size_t)cr0 * (CC / 2) + (kk >> 1)];
      a1 += i2v[(size_t)cr1 * (CC / 2) + (kk >> 1)];
    }
    const v2f* wp = (const v2f*)&sW[(kk >> 1) * 256];
#pragma unroll
    for (int ct = 0; ct < 8; ct++) {
      v2f bf = wp[ct * 16 + lo];
      acc0[ct] = wmma_f32(a0, bf, acc0[ct]);
      acc1[ct] = wmma_f32(a1, bf, acc1[ct]);
    }
  }

#pragma unroll
  for (int ct = 0; ct < 8; ct++) {
    int n = ct * 16 + lo;
    float bs = bias[n];
#pragma unroll
    for (int v = 0; v < 8; v++) {
      int m = v + 8 * hi;
      int row0 = rowBase + wave * 32 + m;
      int row1 = row0 + 16;
      if (row0 < rows) o[(size_t)row0 * CC + n] = fmaxf(acc0[ct][v] + bs, 0.f);
      if (row1 < rows) o[(size_t)row1 * CC + n] = fmaxf(acc1[ct][v] + bs, 0.f);
    }
  }
}

// ---------------- fused layer-3 + policy/value head (K rows only) ---------
// grid: (ceil(K/16), B); one wave per block
__global__ __launch_bounds__(32) void head_kernel(
    const float* __restrict__ H, const float* __restrict__ A,
    const int* __restrict__ def_idx, const int* __restrict__ nop_ptr,
    const float* __restrict__ w1, const float* __restrict__ b1,
    const float* __restrict__ w2, const float* __restrict__ b2,
    const float* __restrict__ pol_w, const float* __restrict__ pol_b,
    const float* __restrict__ val_w, const float* __restrict__ val_b,
    float* __restrict__ pol_out, float* __restrict__ val_out) {
  __shared__ __align__(16) float zT[16 * CC];
  __shared__ __align__(16) float tT[16 * CC];
  int b = blockIdx.y, kt = blockIdx.x;
  int lane = threadIdx.x, lo = lane & 15, hi = lane >> 4;
  const float* Hb = H + (size_t)b * NN * CC;
  const float* Ab = A + (size_t)b * NN * CC;

  // gather z = H[r] + A[r] for 16 defense rows (2 lanes per row)
  {
    int slot = lane & 15, half = lane >> 4;
    int kg = kt * 16 + slot;
    int r = (kg < KK) ? def_idx[b * KK + kg] : 0;
    const float4* hp = (const float4*)&Hb[(size_t)r * CC];
    const float4* ap = (const float4*)&Ab[(size_t)r * CC];
    float4* zp = (float4*)&zT[slot * CC];
    for (int j = half * 16; j < half * 16 + 16; j++) {
      float4 hv = hp[j], av = ap[j];
      zp[j] = make_float4(hv.x + av.x, hv.y + av.y, hv.z + av.z, hv.w + av.w);
    }
  }
  __syncthreads();

  // t = relu(z @ w1 + b1)
  v8f acc[8] = {};
#pragma unroll 4
  for (int k = 0; k < CC; k += 4) {
    int kk = k + 2 * hi;
    v2f a = *(const v2f*)&zT[lo * CC + kk];
#pragma unroll
    for (int ct = 0; ct < 8; ct++) {
      int n = ct * 16 + lo;
      v2f bf;
      bf.x = w1[kk * CC + n];
      bf.y = w1[(kk + 1) * CC + n];
      acc[ct] = wmma_f32(a, bf, acc[ct]);
    }
  }
#pragma unroll
  for (int ct = 0; ct < 8; ct++) {
    int n = ct * 16 + lo;
#pragma unroll
    for (int v = 0; v < 8; v++) {
      int m = v + 8 * hi;
      tT[m * CC + n] = fmaxf(acc[ct][v] + b1[n], 0.f);
    }
  }
  __syncthreads();

  // e = relu(t @ w2 + b2) -> overwrite zT
  v8f acc2[8] = {};
#pragma unroll 4
  for (int k = 0; k < CC; k += 4) {
    int kk = k + 2 * hi;
    v2f a = *(const v2f*)&tT[lo * CC + kk];
#pragma unroll
    for (int ct = 0; ct < 8; ct++) {
      int n = ct * 16 + lo;
      v2f bf;
      bf.x = w2[kk * CC + n];
      bf.y = w2[(kk + 1) * CC + n];
      acc2[ct] = wmma_f32(a, bf, acc2[ct]);
    }
  }
  __syncthreads();
#pragma unroll
  for (int ct = 0; ct < 8; ct++) {
    int n = ct * 16 + lo;
#pragma unroll
    for (int v = 0; v < 8; v++) {
      int m = v + 8 * hi;
      zT[m * CC + n] = fmaxf(acc2[ct][v] + b2[n], 0.f);
    }
  }
  __syncthreads();

  // policy / value heads
  if (lane < 16) {
    int kg = kt * 16 + lane;
    if (kg < KK) {
      float p = pol_b[0];
      for (int c = 0; c < CC; c++) p += zT[lane * CC + c] * pol_w[c];
      pol_out[b * KK + kg] = p;
      if (kg == nop_ptr[0]) {
        float vv = val_b[0];
        for (int c = 0; c < CC; c++) vv += zT[lane * CC + c] * val_w[c];
        val_out[b] = vv;
      }
    }
  }
}

extern "C" void kernel_launch(void* const* d_in, const int* in_sizes, int n_in,
                              void* d_out, int out_size, void* d_ws,
                              size_t ws_size, hipStream_t stream) {
  const float* x        = (const float*)d_in[0];
  const int*   edges    = (const int*)d_in[1];
  const int*   def_idx  = (const int*)d_in[2];
  const int*   nop      = (const int*)d_in[3];
  const float* w1_first = (const float*)d_in[4];
  const float* w1_rest  = (const float*)d_in[5];
  const float* w2       = (const float*)d_in[6];
  const float* b1       = (const float*)d_in[7];
  const float* b2       = (const float*)d_in[8];
  const float* pol_w    = (const float*)d_in[9];
  const float* pol_b    = (const float*)d_in[10];
  const float* val_w    = (const float*)d_in[11];
  const float* val_b    = (const float*)d_in[12];
  float* out = (float*)d_out;

  const size_t S = (size_t)BB * NN * CC;  // floats per activation buffer
  float* T    = (float*)d_ws;
  float* H    = T + S;
  float* A    = H + S;
  float* agg0 = A + S;

  dim3 ggrid((NN + 127) / 128, BB);
  int sc_blocks = (int)(((size_t)BB * EE * 32 + 255) / 256);
  int t0_blocks = (int)(((size_t)BB * NN * 32 + 255) / 256);

  // ---- layer 1 (scalar features) ----
  (void)hipMemsetAsync(agg0, 0, (size_t)BB * NN * sizeof(float), stream);
  scatter0_kernel<<<(BB * EE + 255) / 256, 256, 0, stream>>>(x, edges, agg0);
  t0_kernel<<<t0_blocks, 256, 0, stream>>>(x, agg0, w1_first, b1, T);
  gemm_kernel<false><<<ggrid, 128, 0, stream>>>(T, nullptr, w2, b2, H, NN);

  // ---- layer 2 ----
  (void)hipMemsetAsync(A, 0, S * sizeof(float), stream);
  scatter_kernel<<<sc_blocks, 256, 0, stream>>>(H, edges, A);
  gemm_kernel<true><<<ggrid, 128, 0, stream>>>(H, A, w1_rest, b1 + CC, T, NN);
  gemm_kernel<false><<<ggrid, 128, 0, stream>>>(T, nullptr, w2 + CC * CC,
                                                b2 + CC, H, NN);

  // ---- layer 3: full scatter, but dense math only at K defense rows ----
  (void)hipMemsetAsync(A, 0, S * sizeof(float), stream);
  scatter_kernel<<<sc_blocks, 256, 0, stream>>>(H, edges, A);
  dim3 hgrid((KK + 15) / 16, BB);
  head_kernel<<<hgrid, 32, 0, stream>>>(
      H, A, def_idx, nop, w1_rest + CC * CC, b1 + 2 * CC, w2 + 2 * CC * CC,
      b2 + 2 * CC, pol_w, pol_b, val_w, val_b, out, out + BB * KK);
}